// STFT_47313359732945
// MI455X (gfx1250) — compile-verified
//
#include <hip/hip_runtime.h>
#include <math.h>
#include <stdint.h>

// ---------------------------------------------------------------------------
// STFT magnitude as an fp32 WMMA GEMM on gfx1250 (wave32).
//   out[b, f, t] = sqrt( (basis[f,:]·win)^2 + (basis[f+1025,:]·win)^2 )
// V_WMMA_F32_16X16X4_F32 on the matrix pipe; tiles are staged with CDNA5
// GLOBAL_LOAD_ASYNC_TO_LDS_B128 (direct global->LDS DMA, ASYNCcnt-tracked,
// no VGPR payload -> nothing to spill) into a double-buffered LDS so the
// next stage streams in underneath the WMMA burst.
// ---------------------------------------------------------------------------

typedef __attribute__((ext_vector_type(2))) float v2f;
typedef __attribute__((ext_vector_type(8))) float v8f;

#define FILT 2048      // filter length (K)
#define HOP  512
#define CUT  1025      // output channels (real rows 0..1024, imag rows 1025..2049)
#define NFR  626       // frames per batch
#define SLEN 320000    // samples per batch
#define PAD  1024      // reflect pad each side

#define BM   64        // output channels per WG (4 M-groups x 16)
#define BN   32        // frames per WG (2 N-groups x 16)
#define BK   32        // K chunk per stage
#define LSTR 36        // padded LDS row stride (36 = 4*9, 9 coprime 16 ->
                       //  fragment b64 reads hit 64 distinct banks; 16B aligned)
#define A_SZ (BM * LSTR)                 // one A plane: 2304 floats
#define BUF  (2 * A_SZ + BN * LSTR)      // one buffer:  5760 floats (23 KB)
#define KITERS (FILT / BK)               // 64

__device__ __forceinline__ void async_b128(unsigned ldsOff, const float* g) {
    // global->LDS async DMA: vdst = LDS byte address, vaddr = 64-bit global
    asm volatile("global_load_async_to_lds_b128 %0, %1, off"
                 :: "v"(ldsOff), "v"(g) : "memory");
}

__global__ __launch_bounds__(256, 1)
void stft_mag_wmma(const float* __restrict__ x,
                   const float* __restrict__ basis,
                   float* __restrict__ out)
{
    __shared__ float lds[2 * BUF];        // 46 KB, double buffered

    const int tid  = threadIdx.x;
    const int lane = tid & 31;            // wave32
    const int wave = tid >> 5;            // 8 waves
    const int mg   = wave & 3;            // M group (rows mg*16..)
    const int ng   = wave >> 2;           // N group (frames ng*16..)
    const int b      = blockIdx.z;
    const int f_base = blockIdx.y * BM;
    const int t_base = blockIdx.x * BN;
    const bool boundary = (blockIdx.x == 0) || (blockIdx.x == gridDim.x - 1);

    const float* xb = x + b * SLEN;

    // ---- staging geometry: one b128 slot per lane per 32-row slab (2 per A plane, 1 for B) ---------------
    const int r0  = tid >> 3;             // 0..31: A rows r0 / r0+32; B row r0
    const int c4o = (tid & 7) * 4;        // float col offset (8 f4 per 32-float row)

    const float* gR = basis + (f_base + r0) * FILT + c4o;   // real rows: in-bounds
    int gI0 = f_base + r0 + CUT;       if (gI0 > 2049) gI0 = 2049;
    int gI1 = f_base + r0 + 32 + CUT;  if (gI1 > 2049) gI1 = 2049;
    const unsigned offI0 = (unsigned)(gI0 * FILT + c4o) * 4u;
    const unsigned offI1 = (unsigned)(gI1 * FILT + c4o) * 4u;

    const unsigned ldsBase = (unsigned)(unsigned long long)(uintptr_t)lds;  // addr[31:0]
    const unsigned aOff0 = ldsBase + (unsigned)((r0 * LSTR + c4o) * 4);
    const unsigned bOff0 = ldsBase + (unsigned)((2 * A_SZ + r0 * LSTR + c4o) * 4);
    const int srcBase = (t_base + r0) * HOP + c4o - PAD;     // window sample
    float* bPtr = lds + 2 * A_SZ + r0 * LSTR + c4o;          // boundary path

    // ---- WMMA fragment pointers (ISA 7.12.2) ------------------------------
    const int row16 = lane & 15;
    const int koff  = (lane < 16) ? 0 : 2;
    const float* fragR0 = lds + (mg * 16 + row16) * LSTR + koff;
    const float* fragI0 = lds + A_SZ + (mg * 16 + row16) * LSTR + koff;
    const float* fragB0 = lds + 2 * A_SZ + (ng * 16 + row16) * LSTR + koff;

    v8f accR = {};
    v8f accI = {};

    auto issue = [&](int k0, int p) {     // 4 (boundary) / 5 (interior) asyncs
        const unsigned aOff = aOff0 + (unsigned)(p * (BUF * 4));
        const float* g = gR + k0;
        async_b128(aOff,                               g);
        async_b128(aOff + 32 * LSTR * 4,               g + 32 * FILT);
        async_b128(aOff + A_SZ * 4,
                   (const float*)((const char*)basis + offI0 + 4u * (unsigned)k0));
        async_b128(aOff + A_SZ * 4 + 32 * LSTR * 4,
                   (const float*)((const char*)basis + offI1 + 4u * (unsigned)k0));
        const int src = srcBase + k0;
        if (!boundary) {                  // interior: window is pure global
            async_b128(bOff0 + (unsigned)(p * (BUF * 4)), xb + src);
        } else {                          // reflect edges via VGPR path
            float tmp[4];
            #pragma unroll
            for (int e = 0; e < 4; ++e) {
                int s = src + e;
                if (s < 0) s = -s;
                if (s >= SLEN) s = 2 * SLEN - 2 - s;
                tmp[e] = xb[s];
            }
            *(float4*)(bPtr + p * BUF) = make_float4(tmp[0], tmp[1], tmp[2], tmp[3]);
        }
    };

    issue(0, 0);                          // prologue: stage 0 -> buffer 0
    for (int kt = 0; kt < KITERS; ++kt) {
        const int p = kt & 1;
        if (kt + 1 < KITERS) {
            issue((kt + 1) * BK, p ^ 1);  // stream next stage into other buffer
            // in-order ASYNCcnt: <= (ops just issued) => current stage landed
            if (boundary) asm volatile("s_wait_asynccnt 0x4" ::: "memory");
            else          asm volatile("s_wait_asynccnt 0x5" ::: "memory");
        } else {
            asm volatile("s_wait_asynccnt 0x0" ::: "memory");
        }
        __syncthreads();                  // all waves' stage-kt data visible

        const float* fR = fragR0 + p * BUF;
        const float* fI = fragI0 + p * BUF;
        const float* fB = fragB0 + p * BUF;
        #pragma unroll
        for (int kk = 0; kk < BK; kk += 4) {
            v2f ar = *(const v2f*)(fR + kk);
            v2f ai = *(const v2f*)(fI + kk);
            v2f bb = *(const v2f*)(fB + kk);
            accR = __builtin_amdgcn_wmma_f32_16x16x4_f32(false, ar, false, bb,
                                                         (short)0, accR, false, false);
            accI = __builtin_amdgcn_wmma_f32_16x16x4_f32(false, ai, false, bb,
                                                         (short)0, accI, false, false);
        }
        __syncthreads();                  // buffer p reusable for stage kt+2
    }

    // ---- magnitude + store; C/D layout: VGPR r -> M = (lane<16 ? r : r+8) --
    const int t = t_base + ng * 16 + row16;
    #pragma unroll
    for (int r = 0; r < 8; ++r) {
        int f = f_base + mg * 16 + ((lane < 16) ? r : r + 8);
        if (f < CUT && t < NFR) {
            float re = accR[r];
            float im = accI[r];
            out[(b * CUT + f) * NFR + t] = sqrtf(re * re + im * im);
        }
    }
}

extern "C" void kernel_launch(void* const* d_in, const int* in_sizes, int n_in,
                              void* d_out, int out_size, void* d_ws, size_t ws_size,
                              hipStream_t stream) {
    const float* x     = (const float*)d_in[0];   // (32, 320000) f32
    const float* basis = (const float*)d_in[1];   // (2050, 1, 2048) f32
    float* out         = (float*)d_out;           // (32, 1025, 626) f32

    const int batches = in_sizes[0] / SLEN;       // 32
    dim3 grid((NFR + BN - 1) / BN,                // 20 frame tiles
              (CUT + BM - 1) / BM,                // 17 channel tiles
              batches);
    stft_mag_wmma<<<grid, 256, 0, stream>>>(x, basis, out);
}